// ComputeAllAtomCoords_22874995818830
// MI455X (gfx1250) — compile-verified
//
#include <hip/hip_runtime.h>
#include <math.h>

typedef __attribute__((ext_vector_type(2))) float v2f;
typedef __attribute__((ext_vector_type(8))) float v8f;

#define EPSV 1e-6f

// ---------------------------------------------------------------------------
// Kernel 1: per-residue 4x4 frame chains via V_WMMA_F32_16X16X4_F32.
// One wave32 handles 4 residues. Chain state is P^T held in WMMA B-operand
// layout (2 VGPRs/lane): lane (hi*16 + 4r + j) holds P_r[j][2*hi], P_r[j][2*hi+1].
// Step factor M = RT * rotX is built transposed in the A-operand layout:
// lane (hi*16 + 4r + i) holds M_r[2*hi][i], M_r[2*hi+1][i], computed as a
// branch-free dot product  M[row][j] = dot(RT[row,:], rotX[:,j]).
// D = A x B => diagonal 4x4 blocks are (P_r * M_r)^T; shuffles repack D into
// the B layout for the next step.
// ---------------------------------------------------------------------------
__global__ __launch_bounds__(256)
void frames_kernel(const float* __restrict__ alphas,   // (1,L,10,2)
                   const float* __restrict__ listRs,   // (L,3,3)
                   const float* __restrict__ listTs,   // (L,3)
                   const float4* __restrict__ RTs4,    // (5,10,4,4) as rows
                   const int*   __restrict__ seq,      // (L,)
                   float* __restrict__ outFrames,      // (L,9,4,4)
                   int L)
{
    __shared__ float4 sRT[200];                        // 5*10*4 rows of 4 floats
    for (int i = threadIdx.x; i < 200; i += blockDim.x) sRT[i] = RTs4[i];
    __syncthreads();

    const int lane = threadIdx.x & 31;
    const int wave = blockIdx.x * (blockDim.x >> 5) + (threadIdx.x >> 5);
    if (wave * 4 >= L) return;                         // wave-uniform exit

    const int n  = lane & 15;
    const int hi = lane >> 4;                          // 0: rows/cols 0..1, 1: 2..3
    const int r  = n >> 2;                             // residue slot in wave
    const int j  = n & 3;                              // row (B/state), col (A)
    const int k0 = hi << 1;                            // 0 or 2

    long l = (long)wave * 4 + r;
    if (l >= L) l = L - 1;                             // clamp: keeps EXEC all-ones

    const int sq = seq[l];

    // --- fixed 0/1 lane weights (branch-free selector masks) ---
    const float gx  = (j == 0) ? 1.0f : 0.0f;          // rotX col-j component rows
    const float s1  = (j == 1) ? 1.0f : 0.0f;
    const float s2  = (j == 2) ? 1.0f : 0.0f;
    const float gw  = (j == 3) ? 1.0f : 0.0f;
    const float wlt = (j <  3) ? 1.0f : 0.0f;          // j<3 mask for init
    const float fhi = (float)hi;

    // --- per-lane rotX coefficients: this lane owns joint myk = 2 + j + 4*hi ---
    float ccs, ssn;
    {
        const int myk = 2 + j + (hi << 2);             // 2..9 across the 8 lanes of r
        float a0 = alphas[(l * 10 + myk) * 2 + 0];
        float a1 = alphas[(l * 10 + myk) * 2 + 1];
        float nrm = __builtin_amdgcn_sqrtf(fmaf(a0, a0, a1 * a1)) + EPSV;
        float inv = __builtin_amdgcn_rcpf(nrm);
        ccs = a0 * inv;
        ssn = a1 * inv;
    }

    // --- RTF0 init (arithmetic, no control flow): lane holds P[j][k0..k0+1] ---
    float2 fr[9];
    {
        const int jj = (j < 3) ? j : 2;                // clamped row for safe loads
        float rs0 = listRs[l * 9 + jj * 3 + k0];       // col k0 in {0,2}
        float rs1 = listRs[l * 9 + jj * 3 + (hi ? 2 : 1)];
        float ts  = listTs[l * 3 + jj];
        float c1  = fhi * ts + (1.0f - fhi) * rs1;     // hi? T[j] : R[j][1]
        float b0  = wlt * rs0;                         // j==3 -> 0
        float b1  = wlt * c1 + (1.0f - wlt) * fhi;     // j==3: hi? 1 : 0
        fr[0] = make_float2(b0, b1);
        *(float2*)(outFrames + l * 144 + 0 * 16 + j * 4 + k0) = fr[0];
    }

    const int srcLane = (n < 8) ? n : (n + 16);        // D source lane for repack
    const int rodd = r & 1;
    const int rb4base = sq * 40;                       // float4-row base for this seq

#define STEP(P, K, F) do {                                                      \
    /* broadcast rotX coeffs from the lane owning joint K in this residue */    \
    const int csrc = ((K) < 6) ? ((r << 2) + ((K) - 2))                         \
                               : (16 + (r << 2) + ((K) - 6));                   \
    float cc = __shfl(ccs, csrc, 32);                                           \
    float ss = __shfl(ssn, csrc, 32);                                           \
    /* g = rotX column j (branch-free) */                                       \
    float gy = fmaf(s1, cc, -s2 * ss);                                          \
    float gz = fmaf(s1, ss,  s2 * cc);                                          \
    /* A = (RT*rotX)^T : M[row][j] = dot(RT[row,:], g) for rows k0, k0+1 */     \
    float4 rA = sRT[rb4base + (K) * 4 + k0];                                    \
    float4 rB = sRT[rb4base + (K) * 4 + k0 + 1];                                \
    float m0 = fmaf(gx, rA.x, fmaf(gy, rA.y, fmaf(gz, rA.z, gw * rA.w)));       \
    float m1 = fmaf(gx, rB.x, fmaf(gy, rB.y, fmaf(gz, rB.z, gw * rB.w)));       \
    v2f A; A[0] = m0; A[1] = m1;                                                \
    v2f B; B[0] = fr[P].x; B[1] = fr[P].y;                                      \
    v8f C = {};                                                                 \
    v8f D = __builtin_amdgcn_wmma_f32_16x16x4_f32(false, A, false, B,           \
                                                  (short)0, C, false, false);   \
    /* Repack diagonal blocks of D back into B layout:                          \
       new B[k_row][n] = D[4r + k_row][n]; src vgpr = (4r+k_row)&7,             \
       src lane = n + 16*(r>=2). */                                             \
    float x0 = __shfl(D[0], srcLane, 32);                                       \
    float x2 = __shfl(D[2], srcLane, 32);                                       \
    float x4 = __shfl(D[4], srcLane, 32);                                       \
    float x6 = __shfl(D[6], srcLane, 32);                                       \
    float y1 = __shfl(D[1], srcLane, 32);                                       \
    float y3 = __shfl(D[3], srcLane, 32);                                       \
    float y5 = __shfl(D[5], srcLane, 32);                                       \
    float y7 = __shfl(D[7], srcLane, 32);                                       \
    float nb0 = hi ? (rodd ? x6 : x2) : (rodd ? x4 : x0);                       \
    float nb1 = hi ? (rodd ? y7 : y3) : (rodd ? y5 : y1);                       \
    fr[F] = make_float2(nb0, nb1);                                              \
    *(float2*)(outFrames + l * 144 + (F) * 16 + j * 4 + k0) = fr[F];            \
} while (0)

    STEP(0, 2, 1);
    STEP(1, 3, 2);
    STEP(2, 4, 3);
    STEP(3, 5, 4);
    STEP(3, 6, 5);
    STEP(5, 7, 6);
    STEP(6, 8, 7);
    STEP(6, 9, 8);
#undef STEP
}

// ---------------------------------------------------------------------------
// Kernel 2: per-atom gather + 4x4 matvec. Frames were just written and fit in
// the 192MB L2, so the 64B/thread gather is L2 traffic; stores are contiguous.
// ---------------------------------------------------------------------------
__global__ __launch_bounds__(256)
void xyz_kernel(const float* __restrict__ frames,   // (L,9,4,4)
                const float* __restrict__ xyzBase,  // (5,34,4)
                const int*   __restrict__ seq,      // (L,)
                const int*   __restrict__ baseIdx,  // (5,34)
                float* __restrict__ outXyz,         // (L,34,3)
                int L)
{
    long idx = (long)blockIdx.x * blockDim.x + threadIdx.x;
    long total = (long)L * 34;
    if (idx >= total) return;
    int l = (int)(idx / 34);
    int a = (int)(idx - (long)l * 34);
    int s = seq[l];
    int b = baseIdx[s * 34 + a];
    const float4* P = (const float4*)(frames + (long)l * 144 + b * 16);
    float4 p0 = P[0], p1 = P[1], p2 = P[2];
    float4 v  = *(const float4*)(xyzBase + ((long)s * 34 + a) * 4);
    float x = fmaf(p0.x, v.x, fmaf(p0.y, v.y, fmaf(p0.z, v.z, p0.w * v.w)));
    float y = fmaf(p1.x, v.x, fmaf(p1.y, v.y, fmaf(p1.z, v.z, p1.w * v.w)));
    float z = fmaf(p2.x, v.x, fmaf(p2.y, v.y, fmaf(p2.z, v.z, p2.w * v.w)));
    long o = idx * 3;
    outXyz[o + 0] = x;
    outXyz[o + 1] = y;
    outXyz[o + 2] = z;
}

extern "C" void kernel_launch(void* const* d_in, const int* in_sizes, int n_in,
                              void* d_out, int out_size, void* d_ws, size_t ws_size,
                              hipStream_t stream) {
    const float*  alphas  = (const float*)d_in[0];
    const float*  listRs  = (const float*)d_in[1];
    const float*  listTs  = (const float*)d_in[2];
    const float*  xyzBase = (const float*)d_in[3];
    const float4* RTs4    = (const float4*)d_in[4];
    const int*    seq     = (const int*)d_in[5];
    const int*    baseIdx = (const int*)d_in[6];
    const int L = in_sizes[5];                      // seq has L elements

    float* outFrames = (float*)d_out;               // (1,L,9,4,4) flat
    float* outXyz    = outFrames + (size_t)L * 144; // (1,L,34,3) flat

    int waves   = (L + 3) / 4;                      // 4 residues per wave32
    int blocks1 = (waves + 7) / 8;                  // 8 waves per 256-thread block
    frames_kernel<<<blocks1, 256, 0, stream>>>(alphas, listRs, listTs, RTs4, seq,
                                               outFrames, L);

    long total  = (long)L * 34;
    int blocks2 = (int)((total + 255) / 256);
    xyz_kernel<<<blocks2, 256, 0, stream>>>(outFrames, xyzBase, seq, baseIdx,
                                            outXyz, L);
}